// GraphProjection_74071005986926
// MI455X (gfx1250) — compile-verified
//
#include <hip/hip_runtime.h>

// feat3: 14*14*256 = 50176 floats (200704 B), feat4: 7*7*512 = 25088 floats (100352 B)
#define F3_FLOATS 50176
#define F4_FLOATS 25088
#define F3_CHUNKS (F3_FLOATS * 4 / 16)   // 12544 x 16B
#define F4_CHUNKS (F4_FLOATS * 4 / 16)   // 6272  x 16B
#define LDS_BYTES ((F3_FLOATS + F4_FLOATS) * 4)  // 301056 <= 320KB WGP LDS

struct BL { float w11, w21, w12, w22; int b11, b21, b12, b22; };

__device__ __forceinline__ BL make_bl(float h, float w, float S, int Wd, int C) {
    // reference: x = h / (224/S); S/224 is an exact power-of-two reciprocal here
    float x = h * (S * (1.0f / 224.0f));
    float y = w * (S * (1.0f / 224.0f));
    float x1 = floorf(x), x2 = fminf(ceilf(x), S - 1.0f);
    float y1 = floorf(y), y2 = fminf(ceilf(y), S - 1.0f);
    float dx2 = x2 - x, dx1 = x - x1, dy2 = y2 - y, dy1 = y - y1;
    BL r;
    // Exact reference weight products: all four vanish when floor==ceil (clamped pts)
    r.w11 = dx2 * dy2; r.w21 = dx1 * dy2; r.w12 = dx2 * dy1; r.w22 = dx1 * dy1;
    int xi1 = (int)x1, xi2 = (int)x2, yi1 = (int)y1, yi2 = (int)y2;
    r.b11 = (xi1 * Wd + yi1) * C;
    r.b12 = (xi1 * Wd + yi2) * C;
    r.b21 = (xi2 * Wd + yi1) * C;
    r.b22 = (xi2 * Wd + yi2) * C;
    return r;
}

__device__ __forceinline__ void blend4(const BL& bl, float4 a, float4 b, float4 c, float4 d,
                                       float* __restrict__ o, int c0) {
    __builtin_nontemporal_store(bl.w11*a.x + bl.w21*b.x + bl.w12*c.x + bl.w22*d.x, o + c0 + 0);
    __builtin_nontemporal_store(bl.w11*a.y + bl.w21*b.y + bl.w12*c.y + bl.w22*d.y, o + c0 + 1);
    __builtin_nontemporal_store(bl.w11*a.z + bl.w21*b.z + bl.w12*c.z + bl.w22*d.z, o + c0 + 2);
    __builtin_nontemporal_store(bl.w11*a.w + bl.w21*b.w + bl.w12*c.w + bl.w22*d.w, o + c0 + 3);
}

// C = 64: half-wave, one float4 per active lane.
__device__ __forceinline__ void blend64(const float* __restrict__ f, const BL& bl,
                                        float* __restrict__ o, int lane) {
    const float4* q11 = (const float4*)(f + bl.b11);
    const float4* q21 = (const float4*)(f + bl.b21);
    const float4* q12 = (const float4*)(f + bl.b12);
    const float4* q22 = (const float4*)(f + bl.b22);
    if (lane < 16)
        blend4(bl, q11[lane], q21[lane], q12[lane], q22[lane], o, lane << 2);
}

// C in {128,256,512}: uniform trip count C/128 -> fully unrolled.
template<int C>
__device__ __forceinline__ void blendN(const float* __restrict__ f, const BL& bl,
                                       float* __restrict__ o, int lane) {
    const float4* q11 = (const float4*)(f + bl.b11);
    const float4* q21 = (const float4*)(f + bl.b21);
    const float4* q12 = (const float4*)(f + bl.b12);
    const float4* q22 = (const float4*)(f + bl.b22);
#pragma unroll
    for (int j = 0; j < C / 128; ++j) {
        int c4 = lane + 32 * j;
        blend4(bl, q11[c4], q21[c4], q12[c4], q22[c4], o, c4 << 2);
    }
}

__global__ __launch_bounds__(1024)
void GraphProjection_kernel(const float* __restrict__ coords,
                            const float* __restrict__ f1, const float* __restrict__ f2,
                            const float* __restrict__ f3, const float* __restrict__ f4,
                            float* __restrict__ out, int N) {
    extern __shared__ float lds[];
    float* l3 = lds;
    float* l4 = lds + F3_FLOATS;

    // --- Kick off feat3 + feat4 staging into LDS (CDNA5 async global->LDS b128 DMA).
    //     No wait yet: overlap the 294KB fill with the global-memory passes below. ---
    {
        unsigned l3b = (unsigned)(unsigned long long)(void*)l3;  // low 32 bits = LDS byte offset
        for (unsigned i = threadIdx.x; i < F3_CHUNKS; i += blockDim.x) {
            unsigned go = i * 16u;
            unsigned lo = l3b + go;
            asm volatile("global_load_async_to_lds_b128 %0, %1, %2"
                         :: "v"(lo), "v"(go), "s"(f3) : "memory");
        }
        unsigned l4b = (unsigned)(unsigned long long)(void*)l4;
        for (unsigned i = threadIdx.x; i < F4_CHUNKS; i += blockDim.x) {
            unsigned go = i * 16u;
            unsigned lo = l4b + go;
            asm volatile("global_load_async_to_lds_b128 %0, %1, %2"
                         :: "v"(lo), "v"(go), "s"(f4) : "memory");
        }
    }

    const int lane = threadIdx.x & 31;
    const int wid  = threadIdx.x >> 5;
    const int wpb  = blockDim.x >> 5;
    const int gw   = blockIdx.x * wpb + wid;
    const int nw   = gridDim.x * wpb;

    // --- Pass A: coords + levels 1-2 (global/L2-resident), overlapping the async DMA ---
    for (int p = gw; p < N; p += nw) {
        float X = coords[3 * p + 0];
        float Y = coords[3 * p + 1];
        float Z = coords[3 * p + 2];
        float nz = -Z;
        float h = 250.0f * (-Y) / nz + 112.0f;
        float w = 250.0f * X / nz + 112.0f;
        h = fminf(fmaxf(h, 0.0f), 223.0f);
        w = fminf(fmaxf(w, 0.0f), 223.0f);

        float* orow = out + p * 963;
        if (lane < 3)
            __builtin_nontemporal_store(lane == 0 ? X : (lane == 1 ? Y : Z), orow + lane);

        BL b1 = make_bl(h, w, 56.0f, 56, 64);
        blend64(f1, b1, orow + 3, lane);
        BL b2 = make_bl(h, w, 28.0f, 28, 128);
        blendN<128>(f2, b2, orow + 67, lane);
    }

    // --- DMA completion + workgroup visibility ---
    asm volatile("s_wait_asynccnt 0x0" ::: "memory");
    __syncthreads();

    // --- Pass B: levels 3-4 from LDS (conflict-free ds_load_b128 gathers) ---
    for (int p = gw; p < N; p += nw) {
        float X = coords[3 * p + 0];
        float Y = coords[3 * p + 1];
        float Z = coords[3 * p + 2];
        float nz = -Z;
        float h = 250.0f * (-Y) / nz + 112.0f;
        float w = 250.0f * X / nz + 112.0f;
        h = fminf(fmaxf(h, 0.0f), 223.0f);
        w = fminf(fmaxf(w, 0.0f), 223.0f);

        float* orow = out + p * 963;
        BL b3 = make_bl(h, w, 14.0f, 14, 256);
        blendN<256>(l3, b3, orow + 195, lane);
        BL b4 = make_bl(h, w, 7.0f, 7, 512);
        blendN<512>(l4, b4, orow + 451, lane);
    }
}

extern "C" void kernel_launch(void* const* d_in, const int* in_sizes, int n_in,
                              void* d_out, int out_size, void* d_ws, size_t ws_size,
                              hipStream_t stream) {
    const float* coords = (const float*)d_in[0];
    const float* f1 = (const float*)d_in[1];
    const float* f2 = (const float*)d_in[2];
    const float* f3 = (const float*)d_in[3];
    const float* f4 = (const float*)d_in[4];
    float* out = (float*)d_out;
    int N = in_sizes[0] / 3;

    (void)d_ws; (void)ws_size; (void)n_in; (void)out_size;
    hipFuncSetAttribute((const void*)GraphProjection_kernel,
                        hipFuncAttributeMaxDynamicSharedMemorySize, LDS_BYTES);
    GraphProjection_kernel<<<dim3(1024), dim3(1024), LDS_BYTES, stream>>>(
        coords, f1, f2, f3, f4, out, N);
}